// GMMLayerSlow_61598420959737
// MI455X (gfx1250) — compile-verified
//
#include <hip/hip_runtime.h>
#include <math.h>

typedef __attribute__((ext_vector_type(2))) float v2f;
typedef __attribute__((ext_vector_type(4))) float v4f;
typedef __attribute__((ext_vector_type(8))) float v8f;

#define NPIX   (512 * 512)
#define KCLS   16
#define BATCH  4
#define LOG2PI 1.8378770664093453f
#define EPS    1e-5f

// ---------------------------------------------------------------------------
// zero the moment accumulator (re-run every call: harness does not re-zero ws)
// ---------------------------------------------------------------------------
__global__ void zero_ws_kernel(float* __restrict__ ws, int n) {
    int i = blockIdx.x * blockDim.x + threadIdx.x;
    if (i < n) ws[i] = 0.0f;
}

// ---------------------------------------------------------------------------
// Pass 1: moment reduction as a WMMA GEMM.
//   moments[b][m][n] = sum_p feat_m(pixel p) * Pij[b, class n, p]
//   feat = [1, R, G, B, R^2, RG, RB, G^2, GB, B^2, 0...0]  (16 rows, 10 used)
// Each wave: C(16x16) += A(16x4 feats) * B(4x16 Pij) over its pixel chunk.
// A layout: lane l, VGPR j -> M = l%16, K = j + 2*(l>=16)
// B layout: lane l, VGPR j -> N = l%16, K = j + 2*(l>=16)
// D layout: lane l, VGPR r -> M = r + 8*(l>=16), N = l%16
// ---------------------------------------------------------------------------
__global__ __launch_bounds__(256) void gmm_moments_kernel(
    const float* __restrict__ Pij, const float* __restrict__ I,
    float* __restrict__ moments, int itersPerWave)
{
    const int b    = blockIdx.y;
    const int wave = threadIdx.x >> 5;
    const int lane = threadIdx.x & 31;
    const int m    = lane & 15;   // feature row (A) == class column (B)
    const int half = lane >> 4;

    const float* Pb = Pij + (size_t)b * KCLS * NPIX + (size_t)m * NPIX;
    const float* Ir = I + (size_t)b * 3 * NPIX;
    const float* Ig = Ir + NPIX;
    const float* Ib = Ir + 2 * NPIX;

    // one-hot feature selector weights (branch-free A construction)
    const float w1  = (m == 0) ? 1.f : 0.f;
    const float wr  = (m == 1) ? 1.f : 0.f;
    const float wg  = (m == 2) ? 1.f : 0.f;
    const float wb  = (m == 3) ? 1.f : 0.f;
    const float wrr = (m == 4) ? 1.f : 0.f;
    const float wrg = (m == 5) ? 1.f : 0.f;
    const float wrb = (m == 6) ? 1.f : 0.f;
    const float wgg = (m == 7) ? 1.f : 0.f;
    const float wgb = (m == 8) ? 1.f : 0.f;
    const float wbb = (m == 9) ? 1.f : 0.f;

    const int gw = blockIdx.x * (256 >> 5) + wave;
    const int p0 = gw * itersPerWave * 4;

    v8f acc = {};
    for (int it = 0; it < itersPerWave; ++it) {
        const int p = p0 + it * 4;
        v4f pv = *(const v4f*)(Pb + p);          // Pij[class m, p..p+3]
        const int q = p + 2 * half;
        v2f rv = *(const v2f*)(Ir + q);
        v2f gv = *(const v2f*)(Ig + q);
        v2f bv = *(const v2f*)(Ib + q);

        v2f a;
        {
            float r = rv.x, g = gv.x, bl = bv.x;
            float t1 = fmaf(wrr, r, wr);  t1 = fmaf(wrg, g, t1); t1 = fmaf(wrb, bl, t1);
            float t2 = fmaf(wgg, g, wg);  t2 = fmaf(wgb, bl, t2);
            float t3 = fmaf(wbb, bl, wb);
            a.x = w1 + r * t1 + g * t2 + bl * t3;
        }
        {
            float r = rv.y, g = gv.y, bl = bv.y;
            float t1 = fmaf(wrr, r, wr);  t1 = fmaf(wrg, g, t1); t1 = fmaf(wrb, bl, t1);
            float t2 = fmaf(wgg, g, wg);  t2 = fmaf(wgb, bl, t2);
            float t3 = fmaf(wbb, bl, wb);
            a.y = w1 + r * t1 + g * t2 + bl * t3;
        }
        v2f bm;
        bm.x = half ? pv.z : pv.x;
        bm.y = half ? pv.w : pv.y;

        acc = __builtin_amdgcn_wmma_f32_16x16x4_f32(
            /*neg_a=*/false, a, /*neg_b=*/false, bm,
            /*c_mod=*/(short)0, acc, /*reuse_a=*/false, /*reuse_b=*/false);
    }

    // reduce 8 waves in LDS, then one atomic per (m,n) element per block
    __shared__ float red[8 * 256];
    #pragma unroll
    for (int r = 0; r < 8; ++r)
        red[wave * 256 + (r + 8 * half) * 16 + m] = acc[r];
    __syncthreads();

    const int e = threadIdx.x;
    float s = 0.f;
    #pragma unroll
    for (int w = 0; w < 8; ++w) s += red[w * 256 + e];
    atomicAdd(&moments[b * 256 + e], s);
}

// ---------------------------------------------------------------------------
// Pass 1b: per-(b,k) 3x3 algebra -> mu, sigma^-1 (symmetric 6), c0
//   c0 = log(alpha) - 0.5*(3*log(2pi) + logdet)
// ---------------------------------------------------------------------------
__global__ void gmm_params_kernel(const float* __restrict__ moments,
                                  float* __restrict__ params)
{
    int t = threadIdx.x;
    if (t >= BATCH * KCLS) return;
    int b = t >> 4, k = t & 15;
    const float* M = moments + b * 256;

    float SP  = M[0 * 16 + k];
    float s1r = M[1 * 16 + k], s1g = M[2 * 16 + k], s1b = M[3 * 16 + k];
    float srr = M[4 * 16 + k], srg = M[5 * 16 + k], srb = M[6 * 16 + k];
    float sgg = M[7 * 16 + k], sgb = M[8 * 16 + k], sbb = M[9 * 16 + k];

    float alpha = SP / (float)NPIX;
    float denom = EPS + SP;
    float mur = s1r / denom, mug = s1g / denom, mub = s1b / denom;

    // sig = (S2 - mu*S1^T - S1*mu^T + SP*mu*mu^T)/denom + 1e-3*I
    float a_ = (srr - 2.f * mur * s1r + SP * mur * mur) / denom + 1e-3f;
    float d_ = (sgg - 2.f * mug * s1g + SP * mug * mug) / denom + 1e-3f;
    float f_ = (sbb - 2.f * mub * s1b + SP * mub * mub) / denom + 1e-3f;
    float b_ = (srg - mur * s1g - mug * s1r + SP * mur * mug) / denom;
    float c_ = (srb - mur * s1b - mub * s1r + SP * mur * mub) / denom;
    float e_ = (sgb - mug * s1b - mub * s1g + SP * mug * mub) / denom;

    float cof00 = d_ * f_ - e_ * e_;
    float cof01 = c_ * e_ - b_ * f_;
    float cof02 = b_ * e_ - d_ * c_;
    float cof11 = a_ * f_ - c_ * c_;
    float cof12 = b_ * c_ - a_ * e_;
    float cof22 = a_ * d_ - b_ * b_;
    float det = a_ * cof00 + b_ * cof01 + c_ * cof02;
    float inv = 1.f / det;

    float c0 = logf(alpha) - 0.5f * (3.f * LOG2PI + logf(det));

    float* P = params + t * 12;
    P[0] = mur;          P[1] = mug;          P[2] = mub;
    P[3] = cof00 * inv;  P[4] = cof01 * inv;  P[5] = cof02 * inv;
    P[6] = cof11 * inv;  P[7] = cof12 * inv;  P[8] = cof22 * inv;
    P[9] = c0;           P[10] = 0.f;         P[11] = 0.f;
}

// ---------------------------------------------------------------------------
// Pass 2: E-step.  Qij = exp(c0 - 0.5*quad); normalize over K; stream out.
// ---------------------------------------------------------------------------
__global__ __launch_bounds__(256) void gmm_estep_kernel(
    const float* __restrict__ I, const float* __restrict__ params,
    float* __restrict__ out)
{
    const int b = blockIdx.y;
    __shared__ float pp[KCLS * 12];
    if (threadIdx.x < KCLS * 12) pp[threadIdx.x] = params[b * KCLS * 12 + threadIdx.x];
    __syncthreads();

    const int p = blockIdx.x * 256 + threadIdx.x;
    const float* Ibase = I + (size_t)b * 3 * NPIX;
    float r = Ibase[p], g = Ibase[NPIX + p], bl = Ibase[2 * NPIX + p];

    float qv[KCLS];
    float sum = 0.f;
    #pragma unroll
    for (int k = 0; k < KCLS; ++k) {
        const float* P = pp + k * 12;
        float dx = r - P[0], dy = g - P[1], dz = bl - P[2];
        float quad = P[3] * dx * dx + P[6] * dy * dy + P[8] * dz * dz
                   + 2.f * (P[4] * dx * dy + P[5] * dx * dz + P[7] * dy * dz);
        float q = __expf(P[9] - 0.5f * quad);
        qv[k] = q;
        sum += q;
    }
    float rinv = 1.f / (EPS + sum);
    float* ob = out + (size_t)b * KCLS * NPIX + p;
    #pragma unroll
    for (int k = 0; k < KCLS; ++k) ob[(size_t)k * NPIX] = qv[k] * rinv;
}

// ---------------------------------------------------------------------------
extern "C" void kernel_launch(void* const* d_in, const int* in_sizes, int n_in,
                              void* d_out, int out_size, void* d_ws, size_t ws_size,
                              hipStream_t stream) {
    const float* Pij = (const float*)d_in[0];   // [4,16,512,512]
    const float* I   = (const float*)d_in[1];   // [4,3,512,512]
    float* out = (float*)d_out;                 // [4,16,512,512]

    float* moments = (float*)d_ws;              // BATCH*256 floats
    float* params  = moments + BATCH * 256;     // BATCH*16*12 floats

    zero_ws_kernel<<<dim3(4), dim3(256), 0, stream>>>(moments, BATCH * 256);

    const int blocksPerBatch = 128;                      // 8 waves/block
    const int wavesPerBatch  = blocksPerBatch * 8;       // 1024
    const int itersPerWave   = NPIX / (wavesPerBatch * 4); // 64
    gmm_moments_kernel<<<dim3(blocksPerBatch, BATCH), dim3(256), 0, stream>>>(
        Pij, I, moments, itersPerWave);

    gmm_params_kernel<<<dim3(1), dim3(64), 0, stream>>>(moments, params);

    gmm_estep_kernel<<<dim3(NPIX / 256, BATCH), dim3(256), 0, stream>>>(
        I, params, out);
}